// VariationalGNN_4037269259013
// MI455X (gfx1250) — compile-verified
//
#include <hip/hip_runtime.h>
#include <hip/hip_bf16.h>
#include <math.h>

// Problem constants (from reference setup_inputs)
#define NNODES 20000
#define D      256
#define NPAT   8
#define KC     256          // codes per patient
#define R      272          // padded rows: 17 * 16 (257 valid slots + pad)
#define SCALE  0.0625f      // 1/sqrt(256)
#define LRALPHA 0.1f

typedef float v2f __attribute__((ext_vector_type(2)));
typedef float v8f __attribute__((ext_vector_type(8)));

// ---------------------------------------------------------------------------
// Gather the per-patient dependency cone: rows 0..255 = embed[codes+1],
// row 256 = embed[N-1], rows 257..271 = 0 (padding kept finite).
// ---------------------------------------------------------------------------
__global__ void k_gather(const int* __restrict__ codes,
                         const float* __restrict__ embed,
                         float* __restrict__ X) {
  int row = blockIdx.x, p = blockIdx.y, t = threadIdx.x;
  float v = 0.0f;
  if (row < KC) {
    int n = codes[p * KC + row] + 1;
    v = embed[(size_t)n * D + t];
  } else if (row == KC) {
    v = embed[(size_t)(NNODES - 1) * D + t];
  }
  X[((size_t)p * R + row) * D + t] = v;
}

// ---------------------------------------------------------------------------
// WMMA fp32 GEMM:  Dm[r][c] = sum_k A[r][k] * B(k,c) + bias[c]
//   TRANSB=1: B row-major (cols x K):  B(k,c) = Bm[c*ldb + k]   (A @ W^T)
//   TRANSB=0: B row-major (K x cols):  B(k,c) = Bm[k*ldb + c]   (E @ Wh)
// One wave computes one 16x16 output tile with V_WMMA_F32_16X16X4_F32.
// Block = 256 threads = 8 waves; wave w takes col tile blockIdx.x*8+w.
// ---------------------------------------------------------------------------
template <int TRANSB>
__global__ void k_gemm(const float* __restrict__ A, const float* __restrict__ Bm,
                       const float* __restrict__ bias, float* __restrict__ Dm,
                       int Ktot, int lda, int ldb, int ldc,
                       long strideA, long strideB, long strideD) {
  const int wave = threadIdx.x >> 5;
  const int lane = threadIdx.x & 31;
  const int half = lane >> 4;   // 0: K+{0,1}   1: K+{2,3}
  const int lr   = lane & 15;
  const int p    = blockIdx.z;
  const int row0 = blockIdx.y * 16;
  const int col0 = (blockIdx.x * 8 + wave) * 16;

  const float* Ap = A  + (size_t)p * strideA;
  const float* Bp = Bm + (size_t)p * strideB;
  float*       Dp = Dm + (size_t)p * strideD;

  v8f acc;
  const float bv = bias ? bias[col0 + lr] : 0.0f;
#pragma unroll
  for (int e = 0; e < 8; ++e) acc[e] = bv;

  const float* arow = Ap + (size_t)(row0 + lr) * lda;
  for (int k = 0; k < Ktot; k += 4) {
    const int kk = k + 2 * half;
    v2f a;                                  // A 16x4: lane<16 -> K={k,k+1}, lane>=16 -> K={k+2,k+3}
    a[0] = arow[kk];
    a[1] = arow[kk + 1];
    v2f b;                                  // B 4x16 mirrored layout across halves
    if (TRANSB) {
      const float* br = Bp + (size_t)(col0 + lr) * ldb + kk;
      b[0] = br[0];
      b[1] = br[1];
    } else {
      b[0] = Bp[(size_t)kk * ldb + (col0 + lr)];
      b[1] = Bp[(size_t)(kk + 1) * ldb + (col0 + lr)];
    }
    acc = __builtin_amdgcn_wmma_f32_16x16x4_f32(false, a, false, b,
                                                (short)0, acc, false, false);
  }
  // C/D layout: lane half h, element e -> D[row0 + e + 8h][col0 + lr]
#pragma unroll
  for (int e = 0; e < 8; ++e) {
    Dp[(size_t)(row0 + e + 8 * half) * ldc + (col0 + lr)] = acc[e];
  }
}

// ---------------------------------------------------------------------------
// Attention scores: s_src[j] = Wh[j] . a[0:D], s_dst[j] = Wh[j] . a[D:2D]
// S layout per patient: [0..R) = s_src, [R..2R) = s_dst. Rows > 256 -> 0.
// ---------------------------------------------------------------------------
__global__ void k_scores(const float* __restrict__ WH, const float* __restrict__ avec,
                         float* __restrict__ S) {
  int j = blockIdx.x, p = blockIdx.y, t = threadIdx.x;
  __shared__ float red0[256];
  __shared__ float red1[256];
  float x = (j <= KC) ? WH[((size_t)p * R + j) * D + t] : 0.0f;
  red0[t] = x * avec[t];
  red1[t] = x * avec[D + t];
  __syncthreads();
  for (int s = 128; s > 0; s >>= 1) {
    if (t < s) { red0[t] += red0[t + s]; red1[t] += red1[t + s]; }
    __syncthreads();
  }
  if (t == 0) {
    float* Sp = S + (size_t)p * 2 * R;
    Sp[j]     = red0[0];
    Sp[R + j] = red1[0];
  }
}

// ---------------------------------------------------------------------------
// Normalized attention weights.
//   scored rows j < KA:  E[j][i] = e_ji / sum_i e_ji,
//       e_ji = exp(leaky_relu(s_src[j]+s_dst[i], 0.1) * SCALE) for i < KI, 0 else
//   j == onehot_row: one-hot at column onehot_row (pass-through for node N-1)
//   other rows: all zero. Columns >= KI are exact zeros.
// ---------------------------------------------------------------------------
__global__ void k_attnE(const float* __restrict__ S, float* __restrict__ E,
                        int KI, int onehot_row, int KA) {
  int j = blockIdx.x, p = blockIdx.y, t = threadIdx.x;
  const float* Sp = S + (size_t)p * 2 * R;
  float* Ej = E + ((size_t)p * R + j) * R;
  __shared__ float red[256];

  if (j == onehot_row) {
    for (int i = t; i < R; i += 256) Ej[i] = (i == onehot_row) ? 1.0f : 0.0f;
    return;
  }
  if (j >= KA) {
    for (int i = t; i < R; i += 256) Ej[i] = 0.0f;
    return;
  }
  const float ssrc = Sp[j];
  float ev[2];
  float lsum = 0.0f;
#pragma unroll
  for (int rep = 0; rep < 2; ++rep) {
    int i = t + rep * 256;
    float e = 0.0f;
    if (i < KI) {
      float x = ssrc + Sp[R + i];
      x = (x >= 0.0f ? x : LRALPHA * x) * SCALE;
      e = expf(x);
    }
    ev[rep] = e;
    lsum += e;
  }
  red[t] = lsum;
  __syncthreads();
  for (int s = 128; s > 0; s >>= 1) {
    if (t < s) red[t] += red[t + s];
    __syncthreads();
  }
  const float inv = 1.0f / red[0];
#pragma unroll
  for (int rep = 0; rep < 2; ++rep) {
    int i = t + rep * 256;
    if (i < R) Ej[i] = ev[rep] * inv;
  }
}

// ---------------------------------------------------------------------------
// Row LayerNorm (ddof=1, eps added to std) + activation (0 = ELU, 1 = ReLU).
// ---------------------------------------------------------------------------
__global__ void k_ln_act(float* __restrict__ H, const float* __restrict__ ga,
                         const float* __restrict__ gb, int act) {
  int j = blockIdx.x, p = blockIdx.y, t = threadIdx.x;
  float* row = H + ((size_t)p * R + j) * D;
  __shared__ float red[256];
  float x = row[t];
  red[t] = x;
  __syncthreads();
  for (int s = 128; s > 0; s >>= 1) {
    if (t < s) red[t] += red[t + s];
    __syncthreads();
  }
  const float mean = red[0] * (1.0f / D);
  __syncthreads();
  const float dxm = x - mean;
  red[t] = dxm * dxm;
  __syncthreads();
  for (int s = 128; s > 0; s >>= 1) {
    if (t < s) red[t] += red[t + s];
    __syncthreads();
  }
  const float sd = sqrtf(red[0] * (1.0f / (D - 1)));
  float y = ga[t] * dxm / (sd + 1e-6f) + gb[t];
  if (act == 0) y = (y > 0.0f) ? y : (expf(y) - 1.0f);
  else          y = (y > 0.0f) ? y : 0.0f;
  row[t] = y;
}

// ---------------------------------------------------------------------------
// KLD per patient over the K gathered slots (duplicates included, as in ref):
//   kld = 0.5 * sum_{j<K, c} (exp(lv)-lv-1+mu^2) / K,  mu=HV[:,:256], lv=HV[:,256:]
// ---------------------------------------------------------------------------
__global__ void k_kld(const float* __restrict__ HV, float* __restrict__ out) {
  int p = blockIdx.x, t = threadIdx.x;
  __shared__ float red[256];
  const float* base = HV + (size_t)p * R * 512;
  float acc = 0.0f;
  for (int j = 0; j < KC; ++j) {
    float mu = base[(size_t)j * 512 + t];
    float lv = base[(size_t)j * 512 + 256 + t];
    acc += expf(lv) - lv - 1.0f + mu * mu;
  }
  red[t] = acc;
  __syncthreads();
  for (int s = 128; s > 0; s >>= 1) {
    if (t < s) red[t] += red[t + s];
    __syncthreads();
  }
  if (t == 0) out[p * 2 + 1] = 0.5f * red[0] / (float)KC;
}

// ---------------------------------------------------------------------------
// Final head, one block per patient:
//   u = h2[N-1] @ out_V^T + out_Vb ; w = relu(u @ fc1_W^T + fc1_b)
//   logit = w . fc2_W + fc2_b
// ---------------------------------------------------------------------------
__global__ void k_final(const float* __restrict__ H2,
                        const float* __restrict__ outV, const float* __restrict__ outVb,
                        const float* __restrict__ fc1W, const float* __restrict__ fc1b,
                        const float* __restrict__ fc2W, const float* __restrict__ fc2b,
                        float* __restrict__ out) {
  int p = blockIdx.x, t = threadIdx.x;
  __shared__ float node[256];
  __shared__ float u[256];
  __shared__ float red[256];
  node[t] = H2[((size_t)p * R + KC) * D + t];   // row 256 = node N-1
  __syncthreads();
  float s = 0.0f;
  const float* wv = outV + (size_t)t * D;
  for (int c = 0; c < D; ++c) s += node[c] * wv[c];
  u[t] = s + outVb[t];
  __syncthreads();
  s = 0.0f;
  const float* w1 = fc1W + (size_t)t * D;
  for (int c = 0; c < D; ++c) s += u[c] * w1[c];
  s += fc1b[t];
  s = (s > 0.0f) ? s : 0.0f;
  red[t] = s * fc2W[t];
  __syncthreads();
  for (int st = 128; st > 0; st >>= 1) {
    if (t < st) red[t] += red[t + st];
    __syncthreads();
  }
  if (t == 0) out[p * 2 + 0] = red[0] + fc2b[0];
}

// ---------------------------------------------------------------------------
extern "C" void kernel_launch(void* const* d_in, const int* in_sizes, int n_in,
                              void* d_out, int out_size, void* d_ws, size_t ws_size,
                              hipStream_t stream) {
  (void)in_sizes; (void)n_in; (void)out_size; (void)ws_size;
  const int*   codes   = (const int*)  d_in[0];
  const float* embed   = (const float*)d_in[1];
  const float* in_W    = (const float*)d_in[2];
  const float* in_Wb   = (const float*)d_in[3];
  const float* in_a    = (const float*)d_in[4];
  const float* in_ln_a = (const float*)d_in[5];
  const float* in_ln_b = (const float*)d_in[6];
  const float* p_W     = (const float*)d_in[7];
  const float* p_b     = (const float*)d_in[8];
  const float* out_W   = (const float*)d_in[9];
  const float* out_Wb  = (const float*)d_in[10];
  const float* out_a   = (const float*)d_in[11];
  const float* out_ln_a= (const float*)d_in[12];
  const float* out_ln_b= (const float*)d_in[13];
  const float* out_V   = (const float*)d_in[14];
  const float* out_Vb  = (const float*)d_in[15];
  const float* fc1_W   = (const float*)d_in[16];
  const float* fc1_b   = (const float*)d_in[17];
  const float* fc2_W   = (const float*)d_in[18];
  const float* fc2_b   = (const float*)d_in[19];
  float* out = (float*)d_out;

  // Workspace layout (floats). Total ~13.5 MB -> L2 resident.
  float* ws   = (float*)d_ws;
  const size_t PRD = (size_t)R * D;          // 272*256 per patient
  float* BUF0 = ws;                          // NPAT * R * D
  float* BUF1 = BUF0 + (size_t)NPAT * PRD;
  float* WH   = BUF1 + (size_t)NPAT * PRD;
  float* HV   = WH   + (size_t)NPAT * PRD;   // NPAT * R * 512
  float* Em   = HV   + (size_t)NPAT * R * 512;  // NPAT * R * R
  float* S    = Em   + (size_t)NPAT * R * R;    // NPAT * 2 * R

  const dim3 blk(256);
  const dim3 gRows(R, NPAT);
  const dim3 gGemm(2, R / 16, NPAT);     // 256 output cols
  const dim3 gGemmP(4, R / 16, NPAT);    // 512 output cols (projection)

  // 0) gather dependency cone
  k_gather<<<gRows, blk, 0, stream>>>(codes, embed, BUF0);

  // 1) two input GAT layers
  float* hin  = BUF0;
  float* hout = BUF1;
  for (int l = 0; l < 2; ++l) {
    const float* W  = in_W    + (size_t)l * D * D;
    const float* Wb = in_Wb   + (size_t)l * D;
    const float* av = in_a    + (size_t)l * 2 * D;
    const float* la = in_ln_a + (size_t)l * D;
    const float* lb = in_ln_b + (size_t)l * D;
    k_gemm<1><<<gGemm, blk, 0, stream>>>(hin, W, Wb, WH, D, D, D, D,
                                         (long)PRD, 0L, (long)PRD);
    k_scores<<<gRows, blk, 0, stream>>>(WH, av, S);
    k_attnE<<<gRows, blk, 0, stream>>>(S, Em, KC, KC, KC);   // KI=256, one-hot row 256
    k_gemm<0><<<gGemm, blk, 0, stream>>>(Em, WH, nullptr, hout, R, R, D, D,
                                         (long)R * R, (long)PRD, (long)PRD);
    k_ln_act<<<gRows, blk, 0, stream>>>(hout, la, lb, /*elu*/0);
    float* tmp = hin; hin = hout; hout = tmp;
  }
  // after loop: hin holds h after the input layers

  // 2) variational projection (512-wide), KLD
  k_gemm<1><<<gGemmP, blk, 0, stream>>>(hin, p_W, p_b, HV, D, D, D, 512,
                                        (long)PRD, 0L, (long)R * 512);
  k_kld<<<NPAT, blk, 0, stream>>>(HV, out);

  // 3) output GAT layer on mu = HV[:, :256]  (lda = 512)
  k_gemm<1><<<gGemm, blk, 0, stream>>>(HV, out_W, out_Wb, WH, D, 512, D, D,
                                       (long)R * 512, 0L, (long)PRD);
  k_scores<<<gRows, blk, 0, stream>>>(WH, out_a, S);
  k_attnE<<<gRows, blk, 0, stream>>>(S, Em, KC + 1, -1, KC + 1); // 257 scored slots
  k_gemm<0><<<gGemm, blk, 0, stream>>>(Em, WH, nullptr, BUF1, R, R, D, D,
                                       (long)R * R, (long)PRD, (long)PRD);
  k_ln_act<<<gRows, blk, 0, stream>>>(BUF1, out_ln_a, out_ln_b, /*relu*/1);

  // 4) final MLP head -> logits
  k_final<<<NPAT, blk, 0, stream>>>(BUF1, out_V, out_Vb, fc1_W, fc1_b,
                                    fc2_W, fc2_b, out);
}